// DiscriminativeLoss_36893769072807
// MI455X (gfx1250) — compile-verified
//
#include <hip/hip_runtime.h>
#include <hip/hip_bf16.h>
#include <math.h>

// ---------------------------------------------------------------------------
// Discriminative loss, MI455X (gfx1250, wave32).
// B=8, N=65536, D=32, K=32.  Memory-bound: 2 streaming passes over 64MB of
// embeddings (~130MB -> ~6us at 23.3 TB/s).  Both matmul-shaped reductions
// run on the WMMA path:
//   - segment sums:  sums(K x D) += onehot(K x 4pts) * emb(4pts x D)
//                    via V_WMMA_F32_16X16X4_F32, accumulators in VGPRs,
//                    zero LDS traffic in the streaming loop.
//   - l_dist Gram:   G = mu * mu^T (32x32x32 fp32) per batch.
// ---------------------------------------------------------------------------

#define DELTA_V 0.3f
#define DELTA_D 1.5f
#define GAMMA_W 0.001f

constexpr int KSEG = 32;            // instance ids (== wave32 width)
constexpr int DD   = 32;            // embedding dim
constexpr int NPTS = 65536;         // points per batch
constexpr int BLK_PER_B = 64;       // streaming blocks per batch
constexpr int PTS_PER_BLK = NPTS / BLK_PER_B;   // 1024
constexpr int WAVES = 8;            // 256-thread blocks
constexpr int PTS_PER_WAVE = PTS_PER_BLK / WAVES; // 128 -> 32 groups of 4

typedef __attribute__((ext_vector_type(2))) float v2f;
typedef __attribute__((ext_vector_type(8))) float v8f;

// ---------------------------------------------------------------------------
__global__ void __launch_bounds__(256) zero_kernel(float* p, int n) {
    int i = blockIdx.x * 256 + threadIdx.x;
    if (i < n) p[i] = 0.0f;
}

// ---------------------------------------------------------------------------
// Pass 1: segment sums + counts, GEMM-ified onto WMMA.
// Per wave, per group of 4 points:
//   A(16x4) = one-hot of labels for instance-row tile ti  (2 compares/lane)
//   B(4x16) = emb chunk for dim-col tile dtile            (2 b32 loads/lane)
//   acc[ti][dtile] += A x B   (V_WMMA_F32_16X16X4_F32, exact fp32)
// Counts: instance id == lane id; labels broadcast with v_readlane_b32
// (wave-uniform SGPR, no DS ops / dscnt waits inside the WMMA loop).
__global__ void __launch_bounds__(256) seg_sum_kernel(
        const float* __restrict__ emb, const int* __restrict__ lbl,
        float* __restrict__ g_sums, float* __restrict__ g_counts) {
    __shared__ float acc_s[KSEG * 33];   // [k*33 + d], col 32 = count
    const int tid  = threadIdx.x;
    const int lane = tid & 31;
    const int wave = tid >> 5;
    const int b    = blockIdx.y;

    for (int i = tid; i < KSEG * 33; i += 256) acc_s[i] = 0.0f;
    __syncthreads();

    const float* embb = emb + (size_t)b * NPTS * DD;
    const int*   lblb = lbl + (size_t)b * NPTS;
    const int nw = blockIdx.x * PTS_PER_BLK + wave * PTS_PER_WAVE;

    const int mrow  = lane & 15;          // A row / B col within tile
    const int khalf = (lane >> 4) * 2;    // lanes 16-31 hold K-slices 2,3

    v8f acc00 = {}, acc01 = {}, acc10 = {}, acc11 = {};
    float cnt = 0.0f;

    #pragma unroll 2
    for (int g = 0; g < PTS_PER_WAVE / 4; ++g) {
        const int n = nw + g * 4;
        // labels of this lane's two K-slices (points n+khalf, n+khalf+1)
        const int s_lo = lblb[n + khalf];
        const int s_hi = lblb[n + khalf + 1];

        // B operands: emb rows, coalesced b32 per lane (2x64B per load)
        v2f b0, b1;
        b0.x = embb[(size_t)(n + khalf)     * DD +      mrow];
        b0.y = embb[(size_t)(n + khalf + 1) * DD +      mrow];
        b1.x = embb[(size_t)(n + khalf)     * DD + 16 + mrow];
        b1.y = embb[(size_t)(n + khalf + 1) * DD + 16 + mrow];

        // A operands: one-hot rows (IGNORE=-1 matches nothing -> zero row)
        v2f a0, a1;
        a0.x = (s_lo == mrow)      ? 1.0f : 0.0f;
        a0.y = (s_hi == mrow)      ? 1.0f : 0.0f;
        a1.x = (s_lo == mrow + 16) ? 1.0f : 0.0f;
        a1.y = (s_hi == mrow + 16) ? 1.0f : 0.0f;

        acc00 = __builtin_amdgcn_wmma_f32_16x16x4_f32(false, a0, false, b0,
                                                      (short)0, acc00, false, false);
        acc01 = __builtin_amdgcn_wmma_f32_16x16x4_f32(false, a0, false, b1,
                                                      (short)0, acc01, false, false);
        acc10 = __builtin_amdgcn_wmma_f32_16x16x4_f32(false, a1, false, b0,
                                                      (short)0, acc10, false, false);
        acc11 = __builtin_amdgcn_wmma_f32_16x16x4_f32(false, a1, false, b1,
                                                      (short)0, acc11, false, false);

        // counts: wave-uniform label broadcasts (v_readlane -> SGPR),
        // instance id == lane id.  No LDS/bpermute in the hot loop.
        const int l0 = __builtin_amdgcn_readlane(s_lo, 0);
        const int l1 = __builtin_amdgcn_readlane(s_hi, 0);
        const int l2 = __builtin_amdgcn_readlane(s_lo, 16);
        const int l3 = __builtin_amdgcn_readlane(s_hi, 16);
        cnt += (l0 == lane) ? 1.0f : 0.0f;
        cnt += (l1 == lane) ? 1.0f : 0.0f;
        cnt += (l2 == lane) ? 1.0f : 0.0f;
        cnt += (l3 == lane) ? 1.0f : 0.0f;
    }

    // Flush per-wave register tiles through shared accumulator.
    // C/D layout: VGPR r -> (M = r [lanes 0-15] or r+8 [lanes 16-31], N = lane%16)
    const int roff = (lane >= 16) ? 8 : 0;
    #pragma unroll
    for (int r = 0; r < 8; ++r) {
        const int i0 = r + roff;
        atomicAdd(&acc_s[(i0)      * 33 +      mrow], acc00[r]);
        atomicAdd(&acc_s[(i0)      * 33 + 16 + mrow], acc01[r]);
        atomicAdd(&acc_s[(i0 + 16) * 33 +      mrow], acc10[r]);
        atomicAdd(&acc_s[(i0 + 16) * 33 + 16 + mrow], acc11[r]);
    }
    atomicAdd(&acc_s[lane * 33 + 32], cnt);
    __syncthreads();

    for (int idx = tid; idx < KSEG * 33; idx += 256) {
        const int k = idx / 33, c = idx % 33;
        const float v = acc_s[idx];
        if (c < DD) atomicAdd(&g_sums[((size_t)b * KSEG + k) * DD + c], v);
        else        atomicAdd(&g_counts[b * KSEG + k], v);
    }
}

// ---------------------------------------------------------------------------
// Per-batch center math: mu = sums/counts, musq, l_reg, and l_dist via a
// 32x32x32 fp32 Gram matrix on the WMMA path (V_WMMA_F32_16X16X4_F32).
__global__ void __launch_bounds__(128) centers_kernel(
        const float* __restrict__ g_sums, const float* __restrict__ g_counts,
        float* __restrict__ g_mu, float* __restrict__ g_ldist,
        float* __restrict__ g_lreg) {
    __shared__ float mu_lds[KSEG * DD];
    __shared__ float c_lds[KSEG];
    __shared__ float musq[KSEG];
    __shared__ float red[4];
    __shared__ float ninst_sh, lreg_sh;

    const int tid = threadIdx.x;
    const int b   = blockIdx.x;

    if (tid < KSEG) c_lds[tid] = g_counts[b * KSEG + tid];
    __syncthreads();

    for (int idx = tid; idx < KSEG * DD; idx += 128) {
        const int k = idx >> 5;
        const float m = g_sums[(size_t)b * KSEG * DD + idx] / fmaxf(c_lds[k], 1.0f);
        mu_lds[idx] = m;
        g_mu[(size_t)b * KSEG * DD + idx] = m;
    }
    __syncthreads();

    if (tid < KSEG) {                        // all of wave 0
        float s = 0.0f;
        #pragma unroll
        for (int d = 0; d < DD; ++d) { const float m = mu_lds[tid * DD + d]; s += m * m; }
        musq[tid] = s;
        const float pres = (c_lds[tid] > 0.0f) ? 1.0f : 0.0f;
        float ni = pres;
        float lr = (pres > 0.0f && s > 0.0f) ? sqrtf(s) : 0.0f;
        #pragma unroll
        for (int off = 16; off; off >>= 1) {
            ni += __shfl_xor(ni, off);
            lr += __shfl_xor(lr, off);
        }
        if (tid == 0) { ninst_sh = ni; lreg_sh = lr / fmaxf(ni, 1.0f); }
    }
    __syncthreads();

    // ---- WMMA Gram: G = mu * mu^T, 32x32x32 in fp32 ----
    const int wave = tid >> 5, lane = tid & 31;
    const int ti = wave >> 1, tj = wave & 1;
    const int mrow  = lane & 15;
    const int khalf = (lane >> 4) * 2;
    v8f accv = {};
    #pragma unroll
    for (int kk = 0; kk < DD; kk += 4) {
        v2f a, bb;
        a.x  = mu_lds[(ti * 16 + mrow) * DD + kk + khalf + 0];
        a.y  = mu_lds[(ti * 16 + mrow) * DD + kk + khalf + 1];
        bb.x = mu_lds[(tj * 16 + mrow) * DD + kk + khalf + 0];
        bb.y = mu_lds[(tj * 16 + mrow) * DD + kk + khalf + 1];
        accv = __builtin_amdgcn_wmma_f32_16x16x4_f32(false, a, false, bb,
                                                     (short)0, accv, false, false);
    }

    float local = 0.0f;
    const int j  = tj * 16 + (lane & 15);
    const bool pj = c_lds[j] > 0.0f;
    #pragma unroll
    for (int r = 0; r < 8; ++r) {
        const int i = ti * 16 + r + ((lane >= 16) ? 8 : 0);
        const bool pm = (c_lds[i] > 0.0f) && pj && (i != j);
        const float diff2 = fmaxf(musq[i] + musq[j] - 2.0f * accv[r], 0.0f);
        const float dist  = pm ? sqrtf(diff2) : 0.0f;
        const float h     = fmaxf(2.0f * DELTA_D - dist, 0.0f);
        local += pm ? h * h : 0.0f;
    }
    #pragma unroll
    for (int off = 16; off; off >>= 1) local += __shfl_xor(local, off);
    if (lane == 0) red[wave] = local;
    __syncthreads();

    if (tid == 0) {
        const float ni    = ninst_sh;
        const float numer = red[0] + red[1] + red[2] + red[3];
        const float denom = ni * (ni - 1.0f);
        g_ldist[b] = (ni > 1.0f) ? numer / fmaxf(denom, 1.0f) : 0.0f;
        g_lreg[b]  = lreg_sh;
    }
}

// ---------------------------------------------------------------------------
// Pass 2: per-point hinge distance to own center.  float4 per lane:
// 4 points per wave iteration (global_load_b128), 8-lane 3-shuffle reduce,
// unroll 4 -> 4 independent 512B loads in flight per wave.
__global__ void __launch_bounds__(256) var_kernel(
        const float* __restrict__ emb, const int* __restrict__ lbl,
        const float* __restrict__ g_mu, float* __restrict__ g_hinge) {
    __shared__ float mu_lds[KSEG * DD];
    __shared__ float hacc[WAVES][KSEG];

    const int tid  = threadIdx.x;
    const int wave = tid >> 5;
    const int lane = tid & 31;
    const int b    = blockIdx.y;

    for (int idx = tid; idx < KSEG * DD; idx += 256)
        mu_lds[idx] = g_mu[(size_t)b * KSEG * DD + idx];
    for (int idx = tid; idx < WAVES * KSEG; idx += 256) (&hacc[0][0])[idx] = 0.0f;
    __syncthreads();

    const float* embb = emb + (size_t)b * NPTS * DD;
    const int*   lblb = lbl + (size_t)b * NPTS;
    const int nw = blockIdx.x * PTS_PER_BLK + wave * PTS_PER_WAVE;

    const int p = lane >> 3;       // point within group of 4
    const int q = lane & 7;        // dim-quad within point

    #pragma unroll 4
    for (int g = 0; g < PTS_PER_WAVE / 4; ++g) {
        const int n = nw + g * 4 + p;
        const int s = lblb[n];
        const float4 e = *(const float4*)(embb + (size_t)n * DD + q * 4);
        const int ss = (s >= 0) ? s : 0;
        const float4 m = *(const float4*)(mu_lds + ss * DD + q * 4);
        const float dx = e.x - m.x, dy = e.y - m.y, dz = e.z - m.z, dw = e.w - m.w;
        float v = dx * dx + dy * dy + dz * dz + dw * dw;
        v += __shfl_xor(v, 1);
        v += __shfl_xor(v, 2);
        v += __shfl_xor(v, 4);
        if (q == 0 && s >= 0) {
            const float dist = (v > 0.0f) ? sqrtf(v) : 0.0f;
            const float h = fmaxf(dist - DELTA_V, 0.0f);
            atomicAdd(&hacc[wave][s], h * h);   // 4 point-lanes may collide
        }
    }
    __syncthreads();

    if (tid < KSEG) {
        float v = 0.0f;
        #pragma unroll
        for (int w = 0; w < WAVES; ++w) v += hacc[w][tid];
        atomicAdd(&g_hinge[b * KSEG + tid], v);
    }
}

// ---------------------------------------------------------------------------
__global__ void __launch_bounds__(32) final_kernel(
        const float* __restrict__ g_counts, const float* __restrict__ g_hinge,
        const float* __restrict__ g_ldist, const float* __restrict__ g_lreg,
        float* __restrict__ out, int B) {
    __shared__ float ls[32], vs[32];
    const int t = threadIdx.x;
    float loss = 0.0f, valid = 0.0f;
    if (t < B) {
        float ni = 0.0f, lv = 0.0f;
        for (int k = 0; k < KSEG; ++k) {
            const float c = g_counts[t * KSEG + k];
            if (c > 0.0f) { ni += 1.0f; lv += g_hinge[t * KSEG + k] / c; }
        }
        lv /= fmaxf(ni, 1.0f);
        valid = (ni > 0.0f) ? 1.0f : 0.0f;
        loss  = (lv + g_ldist[t] + GAMMA_W * g_lreg[t]) * valid;
    }
    ls[t] = loss; vs[t] = valid;
    __syncthreads();
    if (t == 0) {
        float sl = 0.0f, sv = 0.0f;
        for (int i = 0; i < 32; ++i) { sl += ls[i]; sv += vs[i]; }
        out[0] = (sv > 0.0f) ? sl / fmaxf(sv, 1.0f) : 0.0f;
    }
}

// ---------------------------------------------------------------------------
extern "C" void kernel_launch(void* const* d_in, const int* in_sizes, int n_in,
                              void* d_out, int out_size, void* d_ws, size_t ws_size,
                              hipStream_t stream) {
    (void)n_in; (void)out_size; (void)ws_size;
    const float* emb = (const float*)d_in[0];
    const int*   lbl = (const int*)d_in[1];     // harness: integer -> const int*
    const int B = in_sizes[1] / NPTS;           // 8

    // Workspace layout (floats): sums | counts | hinge | mu | ldist | lreg
    float* ws       = (float*)d_ws;
    float* g_sums   = ws;                                  // B*K*D
    float* g_counts = g_sums   + (size_t)B * KSEG * DD;    // B*K
    float* g_hinge  = g_counts + (size_t)B * KSEG;         // B*K
    float* g_mu     = g_hinge  + (size_t)B * KSEG;         // B*K*D
    float* g_ldist  = g_mu     + (size_t)B * KSEG * DD;    // B
    float* g_lreg   = g_ldist  + B;                        // B
    const int ztotal = B * (2 * KSEG * DD + 2 * KSEG + 2);

    zero_kernel<<<(ztotal + 255) / 256, 256, 0, stream>>>(ws, ztotal);

    dim3 gstream(BLK_PER_B, B);
    seg_sum_kernel<<<gstream, 256, 0, stream>>>(emb, lbl, g_sums, g_counts);
    centers_kernel<<<B, 128, 0, stream>>>(g_sums, g_counts, g_mu, g_ldist, g_lreg);
    var_kernel<<<gstream, 256, 0, stream>>>(emb, lbl, g_mu, g_hinge);
    final_kernel<<<1, 32, 0, stream>>>(g_counts, g_hinge, g_ldist, g_lreg,
                                       (float*)d_out, B);
}